// BoundaryLoss_4337916969636
// MI455X (gfx1250) — compile-verified
//
#include <hip/hip_runtime.h>
#include <math.h>

// ---------------------------------------------------------------------------
// BoundaryLoss for MI455X (gfx1250).
//
// Rigid transforms preserve distances and dot products, so transform the 6400
// waypoints to GLOBAL frame once and run 1-NN + normal-dot there:
//   score[m,n] = ||p_m||^2 - 2 p_m . w_n     (K=4 GEMM -> V_WMMA_F32_16X16X4_F32)
//   A[m,:] = (p_x, p_y, p_z, ||p||^2)        (M = boundary points)
//   B[:,n] = (-2w_x, -2w_y, -2w_z, 1)        (N = waypoints)
// f32 C/D layout: lane = column (waypoint), VGPR = row (boundary candidate),
// so the argmin over boundary points is a per-lane reduction; one
// shfl_xor(16) merges the two row halves per column at the end.
//
// nn_kernel: 800 single-wave blocks (scalar loop bounds). Each wave owns TWO
// waypoint column tiles -> each A-tile load feeds 2 WMMAs (halves L2 traffic
// and loop overhead per WMMA). A-tiles are streamed through a pointer-
// incremented double buffer (the final +64 prefetch lands in the adjacent
// wop[] allocation: valid memory, value never consumed). Per-tile argmin is
// a v_min3 tree + rarely-taken index-recovery branch.
// ---------------------------------------------------------------------------

typedef float v2f __attribute__((ext_vector_type(2)));
typedef float v8f __attribute__((ext_vector_type(8)));

#define BB      64
#define TT      100
#define NPTS    20000
#define NWP     (BB * TT)        // 6400 waypoints
#define NTILES  (NPTS / 16)      // 1250 boundary tiles (exact)
#define NCOLT   (NWP / 16)       // 400 waypoint column tiles (exact)
#define NCHUNK  4                // split N-loop 4x

__device__ __forceinline__ float min8(const v8f d) {
    return fminf(fminf(fminf(d[0], d[1]), fminf(d[2], d[3])),
                 fminf(fminf(d[4], d[5]), fminf(d[6], d[7])));
}

__device__ __forceinline__ void upd(const v8f d, float m, int t, int rbase,
                                    float& best, int& bidx) {
    if (m < best) {                              // rare divergent branch
        best = m;
        int r = 7;
#pragma unroll
        for (int j = 6; j >= 0; --j)             // descending => lowest tied row wins
            if (d[j] == m) r = j;
        bidx = t * 16 + rbase + r;
    }
}

// --- Kernel 1: pack boundary (x,y,z,|p|^2) and waypoint (-2*w_g, 1) + w_g ---
__global__ void __launch_bounds__(256)
prep_kernel(const float* __restrict__ poses,     // [B,4,4]
            const float* __restrict__ wloc,      // [B,T,3]
            const float* __restrict__ boundary,  // [4,N]
            float* __restrict__ pB,              // [NPTS][4] = x,y,z,p2
            float* __restrict__ wop,             // [NWP][4]  = -2wx,-2wy,-2wz,1
            float* __restrict__ wgl)             // [NWP][4]  = wx,wy,wz,0
{
    int tid = blockIdx.x * blockDim.x + threadIdx.x;
    if (tid < NPTS) {
        float x = boundary[tid];
        float y = boundary[NPTS + tid];
        float z = boundary[2 * NPTS + tid];
        pB[tid * 4 + 0] = x;
        pB[tid * 4 + 1] = y;
        pB[tid * 4 + 2] = z;
        pB[tid * 4 + 3] = x * x + y * y + z * z;
    } else if (tid < NPTS + NWP) {
        int wp = tid - NPTS;
        int b  = wp / TT;
        const float* P = poses + b * 16;          // row-major 4x4
        float wx = wloc[wp * 3 + 0];
        float wy = wloc[wp * 3 + 1];
        float wz = wloc[wp * 3 + 2];
        float gx = P[0] * wx + P[1] * wy + P[2]  * wz + P[3];
        float gy = P[4] * wx + P[5] * wy + P[6]  * wz + P[7];
        float gz = P[8] * wx + P[9] * wy + P[10] * wz + P[11];
        wop[wp * 4 + 0] = -2.0f * gx;
        wop[wp * 4 + 1] = -2.0f * gy;
        wop[wp * 4 + 2] = -2.0f * gz;
        wop[wp * 4 + 3] = 1.0f;
        wgl[wp * 4 + 0] = gx;
        wgl[wp * 4 + 1] = gy;
        wgl[wp * 4 + 2] = gz;
        wgl[wp * 4 + 3] = 0.0f;
    }
}

// --- Kernel 2: WMMA 1-NN. One single-wave block per (column-pair, chunk). --
__global__ void __launch_bounds__(32)
nn_kernel(const float* __restrict__ pB,
          const float* __restrict__ wop,
          float* __restrict__ candVal,   // [NCHUNK][NWP]
          int*   __restrict__ candIdx)   // [NCHUNK][NWP]
{
    // blockIdx-derived => wave-uniform (SGPR) loop bounds.
    const int cp    = blockIdx.x >> 2;           // column pair 0..199
    const int chunk = blockIdx.x & 3;            // 0..3
    const int t0 = (chunk * NTILES) / NCHUNK;
    const int t1 = ((chunk + 1) * NTILES) / NCHUNK;

    const int lane = threadIdx.x & 31;
    const int hi   = lane >> 4;                  // 0: K0/K1, rows 0-7; 1: K2/K3, rows 8-15
    const int mlo  = lane & 15;

    // Two loop-invariant B operands (two waypoint column tiles).
    const int wp0 = (cp * 2 + 0) * 16 + mlo;
    const int wp1 = (cp * 2 + 1) * 16 + mlo;
    const v2f bv0 = *(const v2f*)(wop + wp0 * 4 + (hi << 1));
    const v2f bv1 = *(const v2f*)(wop + wp1 * 4 + (hi << 1));

    // Pointer-incremented per-lane A stream (advances 64 floats per tile).
    const float* ap = pB + mlo * 4 + (hi << 1) + (size_t)t0 * 64;

    float best0 = 3.4e38f, best1 = 3.4e38f;
    int   bidx0 = 0,       bidx1 = 0;
    const int rbase = hi << 3;                   // 0 or 8
    const v8f zero = {};

    v2f av = *(const v2f*)ap;
    for (int t = t0; t < t1; ++t) {
        // Unconditional next-tile prefetch; final one reads into wop[] (safe,
        // never consumed).
        const v2f avn = *(const v2f*)(ap + 64);
        ap += 64;

        // One A-tile feeds two independent WMMAs.
        v8f d0 = __builtin_amdgcn_wmma_f32_16x16x4_f32(
            false, av, false, bv0, (short)0, zero, false, false);
        v8f d1 = __builtin_amdgcn_wmma_f32_16x16x4_f32(
            false, av, false, bv1, (short)0, zero, false, false);

        float m0 = min8(d0);
        float m1 = min8(d1);
        upd(d0, m0, t, rbase, best0, bidx0);
        upd(d1, m1, t, rbase, best1, bidx1);

        av = avn;
    }

    // Merge the two row halves of each column (lane l <-> lane l+16).
    {
        float ob = __shfl_xor(best0, 16, 32);
        int   oi = __shfl_xor(bidx0, 16, 32);
        if (ob < best0 || (ob == best0 && oi < bidx0)) { best0 = ob; bidx0 = oi; }
    }
    {
        float ob = __shfl_xor(best1, 16, 32);
        int   oi = __shfl_xor(bidx1, 16, 32);
        if (ob < best1 || (ob == best1 && oi < bidx1)) { best1 = ob; bidx1 = oi; }
    }

    if (lane < 16) {
        candVal[chunk * NWP + wp0] = best0;
        candIdx[chunk * NWP + wp0] = bidx0;
        candVal[chunk * NWP + wp1] = best1;
        candIdx[chunk * NWP + wp1] = bidx1;
    }
}

// --- Kernel 3: merge chunks, dot with normal, ExpReLU, mean -> d_out[0] ----
__global__ void __launch_bounds__(256)
finish_kernel(const float* __restrict__ candVal,
              const int*   __restrict__ candIdx,
              const float* __restrict__ pB,
              const float* __restrict__ wgl,
              const float* __restrict__ nrm,     // [3,N]
              float* __restrict__ out)
{
    __shared__ float sm[256];
    const int tid = threadIdx.x;
    float acc = 0.0f;
    for (int wp = tid; wp < NWP; wp += 256) {
        float best = 3.4e38f;
        int   bidx = 0;
        for (int c = 0; c < NCHUNK; ++c) {       // ascending idx order => '<' keeps lowest tie
            float v = candVal[c * NWP + wp];
            int   i = candIdx[c * NWP + wp];
            if (v < best) { best = v; bidx = i; }
        }
        float px = pB[bidx * 4 + 0];
        float py = pB[bidx * 4 + 1];
        float pz = pB[bidx * 4 + 2];
        float nx = nrm[bidx];
        float ny = nrm[NPTS + bidx];
        float nz = nrm[2 * NPTS + bidx];
        float wx = wgl[wp * 4 + 0];
        float wy = wgl[wp * 4 + 1];
        float wz = wgl[wp * 4 + 2];
        float dot = (wx - px) * nx + (wy - py) * ny + (wz - pz) * nz;
        // ExpReLU, ALPHA=1, BETA=0.5
        acc += (dot > 0.0f) ? (dot + 1.0f) : expf(0.5f * dot);
    }
    sm[tid] = acc;
    __syncthreads();
    for (int s = 128; s > 0; s >>= 1) {
        if (tid < s) sm[tid] += sm[tid + s];
        __syncthreads();
    }
    if (tid == 0) out[0] = sm[0] / (float)NWP;
}

extern "C" void kernel_launch(void* const* d_in, const int* in_sizes, int n_in,
                              void* d_out, int out_size, void* d_ws, size_t ws_size,
                              hipStream_t stream) {
    const float* poses    = (const float*)d_in[0];  // [B,4,4]
    const float* wloc     = (const float*)d_in[1];  // [B,T,3]
    const float* boundary = (const float*)d_in[2];  // [4,N]
    const float* nrm      = (const float*)d_in[3];  // [3,N]

    char* ws = (char*)d_ws;
    float* pB      = (float*)ws; ws += (size_t)NPTS * 4 * sizeof(float);   // 320 KB
    float* wop     = (float*)ws; ws += (size_t)NWP  * 4 * sizeof(float);   // 100 KB (also absorbs
                                                                           //  the final A prefetch)
    float* wgl     = (float*)ws; ws += (size_t)NWP  * 4 * sizeof(float);   // 100 KB
    float* candVal = (float*)ws; ws += (size_t)NCHUNK * NWP * sizeof(float);
    int*   candIdx = (int*)ws;

    prep_kernel<<<(NPTS + NWP + 255) / 256, 256, 0, stream>>>(
        poses, wloc, boundary, pB, wop, wgl);

    // 200 column pairs * 4 chunks = 800 single-wave blocks.
    nn_kernel<<<(NCOLT / 2) * NCHUNK, 32, 0, stream>>>(
        pB, wop, candVal, candIdx);

    finish_kernel<<<1, 256, 0, stream>>>(
        candVal, candIdx, pB, wgl, nrm, (float*)d_out);
}